// VectorQuantizer_39453569581549
// MI455X (gfx1250) — compile-verified
//
#include <hip/hip_runtime.h>

typedef __attribute__((ext_vector_type(2))) float v2f;
typedef __attribute__((ext_vector_type(8))) float v8f;

#define K_CODES 1024
#define D_CH 64
#define HW 4096
#define N_ROWS 131072
#define EMB_STRIDE 68            // 64 + 4 pad: conflict-free b64 B-fragment loads
#define WAVES_PER_WG 8
#define NUM_WGS 256
#define TILES_PER_WAVE (N_ROWS / 16 / (WAVES_PER_WG * NUM_WGS))  // = 4

__global__ void vq_init_loss(float* loss) { *loss = 0.0f; }

__global__ __launch_bounds__(256)
void vq_main(const float* __restrict__ z, const float* __restrict__ emb,
             float* __restrict__ out_zq, float* __restrict__ out_idx,
             float* __restrict__ out_loss) {
  extern __shared__ float smem[];
  float* emb_s = smem;                               // 1024 * 68 floats
  float* esq_s = smem + K_CODES * EMB_STRIDE;        // 1024 floats
  int*   wsc_a = (int*)(esq_s + K_CODES);            // 8 waves * 16 codes

  const int tid = threadIdx.x;

  // ---- stage embedding into LDS (row stride 68), fully coalesced b128 loads
  for (int c = tid; c < K_CODES * 16; c += 256) {
    const int r = c >> 4;
    const int q = (c & 15) << 2;
    const float4 v = reinterpret_cast<const float4*>(emb)[c];
    *reinterpret_cast<float4*>(&emb_s[r * EMB_STRIDE + q]) = v;
  }
  __syncthreads();
  // ---- e_sq[k] = |e_k|^2
  for (int k = tid; k < K_CODES; k += 256) {
    float s = 0.0f;
#pragma unroll 8
    for (int d = 0; d < D_CH; ++d) {
      const float e = emb_s[k * EMB_STRIDE + d];
      s += e * e;
    }
    esq_s[k] = s;
  }
  __syncthreads();

  const int wave = tid >> 5;
  const int lane = tid & 31;
  const int m    = lane & 15;    // row (A) / code col (B,C)
  const int kg   = lane >> 4;    // half-wave group
  int* wsc = wsc_a + wave * 16;

  float lacc = 0.0f;
  const int gw = blockIdx.x * WAVES_PER_WG + wave;

  for (int ti = 0; ti < TILES_PER_WAVE; ++ti) {
    const int tile = gw * TILES_PER_WAVE + ti;
    const int n0 = tile << 4;
    const int b  = n0 >> 12;               // 16-row tile never crosses batch
    const int hw = n0 & (HW - 1);
    const float* zb = z + b * (D_CH * HW) + hw + m;

    // ---- A fragments: 16 k-steps of fp32 16x16x4 layout, kept in VGPRs
    v2f a[16];
#pragma unroll
    for (int s = 0; s < 16; ++s) {
      const int d0 = 4 * s + 2 * kg;
      a[s].x = zb[d0 * HW];
      a[s].y = zb[(d0 + 1) * HW];
    }

    float best[8];
    int   bidx[8];
#pragma unroll
    for (int r = 0; r < 8; ++r) { best[r] = 3.4e38f; bidx[r] = 0; }

    // ---- sweep 64 code tiles: 16 WMMAs each (K=64), fused score + argmin
    for (int t = 0; t < K_CODES / 16; ++t) {
      const float* bp = &emb_s[(t * 16 + m) * EMB_STRIDE + 2 * kg];
      v8f c = {};
#pragma unroll
      for (int s = 0; s < 16; ++s) {
        const v2f bf = *reinterpret_cast<const v2f*>(bp + 4 * s);
        c = __builtin_amdgcn_wmma_f32_16x16x4_f32(
                false, a[s], false, bf, (short)0, c, false, false);
      }
      const float es   = esq_s[t * 16 + m];
      const int   code = t * 16 + m;
#pragma unroll
      for (int r = 0; r < 8; ++r) {
        const float sc = es - 2.0f * c[r];   // |e|^2 - 2 z.e  (|z|^2 const/row)
        if (sc < best[r]) { best[r] = sc; bidx[r] = code; }
      }
    }

    // ---- argmin across each 16-lane group (rows r / r+8), low-index tiebreak
#pragma unroll
    for (int off = 1; off < 16; off <<= 1) {
#pragma unroll
      for (int r = 0; r < 8; ++r) {
        const float os = __shfl_xor(best[r], off, 16);
        const int   oi = __shfl_xor(bidx[r], off, 16);
        if (os < best[r] || (os == best[r] && oi < bidx[r])) {
          best[r] = os; bidx[r] = oi;
        }
      }
    }
#pragma unroll
    for (int r = 0; r < 8; ++r) {
      if (m == 0) {
        const int row = (kg << 3) + r;
        wsc[row] = bidx[r];
        out_idx[n0 + row] = (float)bidx[r];
      }
    }

    // ---- gather code vector, write z_q (coalesced), accumulate loss
    const int code = wsc[m];                 // per-wave LDS, DS is in-order
    float* ob = out_zq + b * (D_CH * HW) + hw + m;
#pragma unroll 8
    for (int i = 0; i < 32; ++i) {
      const int d = 2 * i + kg;
      const float e  = emb_s[code * EMB_STRIDE + d];
      const float zv = zb[d * HW];
      const float df = zv - e;
      lacc += df * df;
      ob[d * HW] = e;
    }
  }

  // ---- loss: vq = (1 + 0.25) * mean((z - z_q)^2)
#pragma unroll
  for (int off = 16; off; off >>= 1) lacc += __shfl_xor(lacc, off, 32);
  if (lane == 0) atomicAdd(out_loss, lacc * (1.25f / 8388608.0f));
}

extern "C" void kernel_launch(void* const* d_in, const int* in_sizes, int n_in,
                              void* d_out, int out_size, void* d_ws, size_t ws_size,
                              hipStream_t stream) {
  const float* z   = (const float*)d_in[0];   // (32, 64, 64, 64)
  const float* emb = (const float*)d_in[1];   // (1024, 64)
  float* out      = (float*)d_out;
  float* out_zq   = out;                       // 8388608 floats
  float* out_idx  = out + 8388608;             // 131072 floats
  float* out_loss = out + 8388608 + 131072;    // 1 float

  vq_init_loss<<<1, 1, 0, stream>>>(out_loss);

  const size_t shmem =
      (size_t)(K_CODES * EMB_STRIDE + K_CODES) * sizeof(float) +
      (size_t)(WAVES_PER_WG * 16) * sizeof(int);
  vq_main<<<NUM_WGS, 256, shmem, stream>>>(z, emb, out_zq, out_idx, out_loss);
}